// bem_block_80917183857443
// MI455X (gfx1250) — compile-verified
//
#include <hip/hip_runtime.h>
#include <math.h>

typedef float v2f __attribute__((ext_vector_type(2)));
typedef float v8f __attribute__((ext_vector_type(8)));

#define Bb 4
#define Nn 16000
#define Kk 24
#define C2c 16
#define NT 16
#define NWAVES 4
#define NTHREADS 128
#define KN (Kk * Nn)

static __device__ __forceinline__ v8f wmma4(v2f a, v2f b, v8f c) {
  // D = A(16x4 f32) * B(4x16 f32) + C(16x16 f32)
  return __builtin_amdgcn_wmma_f32_16x16x4_f32(false, a, false, b, (short)0, c,
                                               false, false);
}

__global__ __launch_bounds__(NTHREADS) void fused_point_attn(
    const float* __restrict__ xyz,   // [B,N,9]
    const float* __restrict__ feat,  // [B,16,N]
    const int* __restrict__ nidx,    // [B,K*N]
    const float* __restrict__ w1, const float* __restrict__ b1,
    const float* __restrict__ w2, const float* __restrict__ b2,
    const float* __restrict__ w3, const float* __restrict__ b3,
    const float* __restrict__ w4, const float* __restrict__ b4,
    const float* __restrict__ w5, const float* __restrict__ b5,
    const float* __restrict__ w6, const float* __restrict__ b6,
    const float* __restrict__ w7, const float* __restrict__ b7,
    float* __restrict__ out) {
  // ---- LDS ----
  // Weight K-dims padded/permuted to multiples of 4 so fragments line up.
  // K-permutation for W1/W4: cols 0-8 = in 0-8, cols 12-20 = in 9-17, rest 0.
  __shared__ float sW1[32][24];  // w1 18x18 (rows 18-31 zero)
  __shared__ float sW2[32][32];  // w2 32x32
  __shared__ float sW3[16][32];  // w3 9x32  (rows 9-15 zero)
  __shared__ float sW4[16][24];  // w4 16x18
  __shared__ float sW5[16][28];  // w5 3x27  (rows 3-15, col 27 zero)
  __shared__ float sW6[16][48];  // w6 3x48  (rows 3-15 zero)
  __shared__ float sW7[3][3];
  __shared__ float sB1[18], sB2[32], sB3[9], sB4[16], sB5[3], sB6[3], sB7[3];
  // Per-wave operand / result panels (16 pairs each):
  __shared__ float sX1[NWAVES][24][NT];  // [CX 0-8 | 0 | NX 0-8 | 0]
  __shared__ float sX2[NWAVES][32][NT];  // [CF 0-15 | NF 0-15]
  __shared__ float sP5[NWAVES][28][NT];  // [sx 0-8 | pij 0-17 | 0]
  __shared__ float sP6[NWAVES][48][NT];  // [sf 0-15 | fij 0-31]
  // Block-shared reduction inputs:
  __shared__ float gLpi[3][Kk][NT], gLfi[3][Kk][NT];

  const int tid = threadIdx.x;
  const int b = blockIdx.y;
  const int n0 = blockIdx.x * NT;

  // ---- stage weights (zero-padded / K-permuted); shift-only indexing ----
  {
    // sW1: 32 rows x 24 cols; 4 threads per row, 6 cols each
    int o = tid >> 2, iq = (tid & 3) * 6;
#pragma unroll
    for (int u = 0; u < 6; ++u) {
      int i = iq + u;
      float v = 0.f;
      if (o < 18) {
        if (i < 9) v = w1[o * 18 + i];
        else if (i >= 12 && i < 21) v = w1[o * 18 + i - 3];
      }
      sW1[o][i] = v;
    }
  }
  {
    int o = tid >> 2, iq = (tid & 3) * 8;  // sW2: 32x32
#pragma unroll
    for (int u = 0; u < 8; ++u) sW2[o][iq + u] = w2[o * 32 + iq + u];
  }
  if (tid < 128) {
    int o = tid >> 3, iq = (tid & 7) * 4;  // sW3: 16x32
#pragma unroll
    for (int u = 0; u < 4; ++u) {
      int i = iq + u;
      sW3[o][i] = (o < 9) ? w3[o * 32 + i] : 0.f;
    }
  }
  {
    int o = tid >> 3, iq = (tid & 7) * 3;  // sW4: 16x24
#pragma unroll
    for (int u = 0; u < 3; ++u) {
      int i = iq + u;
      float v = 0.f;
      if (i < 9) v = w4[o * 18 + i];
      else if (i >= 12 && i < 21) v = w4[o * 18 + i - 3];
      sW4[o][i] = v;
    }
  }
  {
    int o = tid >> 3, iq = (tid & 7) * 4;  // sW5: 16x28
#pragma unroll
    for (int u = 0; u < 4; ++u) {
      int i = iq + u;
      if (i < 28) sW5[o][i] = (o < 3 && i < 27) ? w5[o * 27 + i] : 0.f;
    }
  }
  {
    int o = tid >> 3, iq = (tid & 7) * 6;  // sW6: 16x48
#pragma unroll
    for (int u = 0; u < 6; ++u) {
      int i = iq + u;
      sW6[o][i] = (o < 3) ? w6[o * 48 + i] : 0.f;
    }
  }
  if (tid < 9) sW7[tid / 3][tid % 3] = w7[tid];
  if (tid < 18) sB1[tid] = b1[tid];
  if (tid < 32) sB2[tid] = b2[tid];
  if (tid < 9) sB3[tid] = b3[tid];
  if (tid < 16) sB4[tid] = b4[tid];
  if (tid < 3) {
    sB5[tid] = b5[tid];
    sB6[tid] = b6[tid];
    sB7[tid] = b7[tid];
  }

  const int w = tid >> 5;
  const int lane = tid & 31;
  const int col = lane & 15;  // output column (n-local) == A-fragment row m
  const int hi = lane >> 4;
  const int kh = 2 * hi;  // K sub-offset of this half-wave inside a fragment

  // ---- per-wave: iteration-invariant halves of X1/X2 (centers) + pads ----
  for (int e = lane; e < 24 * NT; e += 32) {
    int r = e >> 4, c = e & 15;
    sX1[w][r][c] = (r < 9) ? xyz[(b * Nn + n0 + c) * 9 + r] : 0.f;
  }
  for (int e = lane; e < 32 * NT; e += 32) {
    int r = e >> 4, c = e & 15;
    sX2[w][r][c] = (r < 16) ? feat[(b * C2c + r) * Nn + n0 + c] : 0.f;
  }
  if (lane < NT) sP5[w][27][lane] = 0.f;
  __syncthreads();

  // ---- per-lane loop-invariant bias registers (row = v + 8*hi) ----
  float b1r[8], b2r[8], b2s[8], b3r[8], b4r[8], b5r[3], b6r[3];
#pragma unroll
  for (int v = 0; v < 8; ++v) {
    int row = v + 8 * hi;
    b1r[v] = sB1[9 + row - 9];  // sB1[row]
    b2r[v] = sB2[row];
    b2s[v] = sB2[16 + row];
    b4r[v] = sB4[row];
  }
  b3r[0] = sB3[8 * hi];  // v=0 row is 0 (hi=0) or 8 (hi=1)
#pragma unroll
  for (int v = 1; v < 8; ++v) b3r[v] = sB3[v];  // only hi==0 uses v>=1
  const float b1t0 = sB1[16], b1t1 = sB1[17];
#pragma unroll
  for (int v = 0; v < 3; ++v) {
    b5r[v] = sB5[v];
    b6r[v] = sB6[v];
  }

  float* outShift = out + Bb * 6 * Nn;  // second return value, [B,9,K*N]
  float* shiftBase = outShift + b * 9 * KN + n0 + col;

  for (int it = 0; it < 6; ++it) {
    const int k = w * 6 + it;
    // ---- gather neighbor vectors for this wave's 16 (k,n) pairs ----
    const int j = nidx[b * KN + k * Nn + n0 + col];
    if (hi == 0) {
#pragma unroll
      for (int d = 0; d < 9; ++d)
        sX1[w][12 + d][col] = xyz[(b * Nn + j) * 9 + d];
    } else {
#pragma unroll
      for (int f = 0; f < 16; ++f)
        sX2[w][16 + f][col] = feat[(b * C2c + f) * Nn + j];
    }

    // ---- batch-load B fragments + per-pair addends (overlap DS latency) ----
    v2f bf1[6], bf2[8];
#pragma unroll
    for (int s = 0; s < 6; ++s) {
      bf1[s].x = sX1[w][4 * s + kh][col];
      bf1[s].y = sX1[w][4 * s + kh + 1][col];
    }
#pragma unroll
    for (int s = 0; s < 8; ++s) {
      bf2[s].x = sX2[w][4 * s + kh][col];
      bf2[s].y = sX2[w][4 * s + kh + 1][col];
    }
    float nxadd[8], nfadd[8];
    nxadd[0] = sX1[w][12 + 8 * hi][col];  // row 0 / 8
#pragma unroll
    for (int v = 1; v < 8; ++v) nxadd[v] = sX1[w][12 + v][col];
#pragma unroll
    for (int v = 0; v < 8; ++v) nfadd[v] = sX2[w][16 + v + 8 * hi][col];

    // ---- GEMM1: pij(18x16) = W1 @ [tile_xyz ; neigh_xyz] ----
    {
      v8f a0c = {0.f, 0.f, 0.f, 0.f, 0.f, 0.f, 0.f, 0.f};
      v8f a1c = {0.f, 0.f, 0.f, 0.f, 0.f, 0.f, 0.f, 0.f};
#pragma unroll
      for (int s = 0; s < 6; ++s) {
        v2f a0, a1;
        a0.x = sW1[col][4 * s + kh];
        a0.y = sW1[col][4 * s + kh + 1];
        a1.x = sW1[16 + col][4 * s + kh];
        a1.y = sW1[16 + col][4 * s + kh + 1];
        a0c = wmma4(a0, bf1[s], a0c);
        a1c = wmma4(a1, bf1[s], a1c);
      }
#pragma unroll
      for (int v = 0; v < 8; ++v)
        sP5[w][9 + v + 8 * hi][col] = a0c[v] + b1r[v];
      if (hi == 0) {
        sP5[w][25][col] = a1c[0] + b1t0;
        sP5[w][26][col] = a1c[1] + b1t1;
      }
    }
    // ---- GEMM4: sf(16x16) = W4 @ [neigh_xyz ; tile-neigh]; B from bf1 ----
    {
      v8f acc = {0.f, 0.f, 0.f, 0.f, 0.f, 0.f, 0.f, 0.f};
#pragma unroll
      for (int s = 0; s < 6; ++s) {
        v2f a, bf;
        a.x = sW4[col][4 * s + kh];
        a.y = sW4[col][4 * s + kh + 1];
        if (s < 3) {
          bf = bf1[s + 3];
        } else {
          bf.x = bf1[s - 3].x - bf1[s].x;
          bf.y = bf1[s - 3].y - bf1[s].y;
        }
        acc = wmma4(a, bf, acc);
      }
#pragma unroll
      for (int v = 0; v < 8; ++v)
        sP6[w][v + 8 * hi][col] = acc[v] + b4r[v] + nfadd[v];
    }
    // ---- GEMM2: fij(32x16) = W2 @ [tile_feat ; neigh_feat] ----
    {
      v8f a0c = {0.f, 0.f, 0.f, 0.f, 0.f, 0.f, 0.f, 0.f};
      v8f a1c = {0.f, 0.f, 0.f, 0.f, 0.f, 0.f, 0.f, 0.f};
#pragma unroll
      for (int s = 0; s < 8; ++s) {
        v2f a0, a1;
        a0.x = sW2[col][4 * s + kh];
        a0.y = sW2[col][4 * s + kh + 1];
        a1.x = sW2[16 + col][4 * s + kh];
        a1.y = sW2[16 + col][4 * s + kh + 1];
        a0c = wmma4(a0, bf2[s], a0c);
        a1c = wmma4(a1, bf2[s], a1c);
      }
#pragma unroll
      for (int v = 0; v < 8; ++v) {
        sP6[w][16 + v + 8 * hi][col] = a0c[v] + b2r[v];
        sP6[w][32 + v + 8 * hi][col] = a1c[v] + b2s[v];
      }
    }
    // ---- GEMM3: sx(9x16) = W3 @ [neigh_feat ; tile-neigh]; B from bf2 ----
    {
      v8f acc = {0.f, 0.f, 0.f, 0.f, 0.f, 0.f, 0.f, 0.f};
#pragma unroll
      for (int s = 0; s < 8; ++s) {
        v2f a, bf;
        a.x = sW3[col][4 * s + kh];
        a.y = sW3[col][4 * s + kh + 1];
        if (s < 4) {
          bf = bf2[s + 4];
        } else {
          bf.x = bf2[s - 4].x - bf2[s].x;
          bf.y = bf2[s - 4].y - bf2[s].y;
        }
        acc = wmma4(a, bf, acc);
      }
      float* sb = shiftBase + k * Nn;
      // v = 0: valid on ALL lanes (row 0 for hi=0, row 8 for hi=1)
      {
        int row0 = 8 * hi;
        float val = acc[0] + b3r[0] + nxadd[0];
        sP5[w][row0][col] = val;
        sb[row0 * KN] = val;
      }
      if (hi == 0) {  // v = 1..7 -> rows 1..7, hi==0 half only
#pragma unroll
        for (int v = 1; v < 8; ++v) {
          float val = acc[v] + b3r[v] + nxadd[v];
          sP5[w][v][col] = val;
          sb[v * KN] = val;
        }
      }
    }
    // ---- GEMM5: lpi(3x16) = W5 @ [sx ; pij] (P5, 28x16) ----
    {
      v8f acc = {0.f, 0.f, 0.f, 0.f, 0.f, 0.f, 0.f, 0.f};
#pragma unroll
      for (int s = 0; s < 7; ++s) {
        v2f a, bf;
        a.x = sW5[col][4 * s + kh];
        a.y = sW5[col][4 * s + kh + 1];
        bf.x = sP5[w][4 * s + kh][col];
        bf.y = sP5[w][4 * s + kh + 1][col];
        acc = wmma4(a, bf, acc);
      }
      if (hi == 0) {
#pragma unroll
        for (int v = 0; v < 3; ++v) gLpi[v][k][col] = acc[v] + b5r[v];
      }
    }
    // ---- GEMM6: lfi(3x16) = W6 @ [sf ; fij] (P6, 48x16) ----
    {
      v8f acc = {0.f, 0.f, 0.f, 0.f, 0.f, 0.f, 0.f, 0.f};
#pragma unroll
      for (int s = 0; s < 12; ++s) {
        v2f a, bf;
        a.x = sW6[col][4 * s + kh];
        a.y = sW6[col][4 * s + kh + 1];
        bf.x = sP6[w][4 * s + kh][col];
        bf.y = sP6[w][4 * s + kh + 1][col];
        acc = wmma4(a, bf, acc);
      }
      if (hi == 0) {
#pragma unroll
        for (int v = 0; v < 3; ++v) gLfi[v][k][col] = acc[v] + b6r[v];
      }
    }
  }
  __syncthreads();

  // ---- attn = W7@lfi, softmax over k-triples (view quirk), sum/max ----
  if (tid < NT * 3) {
    const int nl = tid / 3, m = tid % 3;
    float w7r[3][3], b7r[3];
#pragma unroll
    for (int c = 0; c < 3; ++c) {
      b7r[c] = sB7[c];
#pragma unroll
      for (int cc = 0; cc < 3; ++cc) w7r[c][cc] = sW7[c][cc];
    }
    float api = 0.f, afi = -1e30f;
    for (int j8 = 0; j8 < 8; ++j8) {
      const int kb = 3 * j8;
      float lf[3][3];
#pragma unroll
      for (int cc = 0; cc < 3; ++cc)
#pragma unroll
        for (int t = 0; t < 3; ++t) lf[cc][t] = gLfi[cc][kb + t][nl];
#pragma unroll
      for (int c = 0; c < 3; ++c) {
        float a0 = b7r[c], a1 = b7r[c], a2 = b7r[c];
#pragma unroll
        for (int cc = 0; cc < 3; ++cc) {
          a0 += w7r[c][cc] * lf[cc][0];
          a1 += w7r[c][cc] * lf[cc][1];
          a2 += w7r[c][cc] * lf[cc][2];
        }
        float mx = fmaxf(a0, fmaxf(a1, a2));
        float e0 = __expf(a0 - mx), e1 = __expf(a1 - mx), e2 = __expf(a2 - mx);
        float inv = 1.f / (e0 + e1 + e2);
        float em = (m == 0) ? e0 : ((m == 1) ? e1 : e2);
        api += em * inv * gLpi[c][kb + m][nl];
        afi = fmaxf(afi, lf[c][m]);
      }
    }
    out[(b * 6 + m) * Nn + n0 + nl] = api;
    out[(b * 6 + 3 + m) * Nn + n0 + nl] = afi;
  }
}

extern "C" void kernel_launch(void* const* d_in, const int* in_sizes, int n_in,
                              void* d_out, int out_size, void* d_ws,
                              size_t ws_size, hipStream_t stream) {
  const float* xyz = (const float*)d_in[0];
  const float* feat = (const float*)d_in[1];
  const int* nidx = (const int*)d_in[2];
  const float* w1 = (const float*)d_in[3];
  const float* b1 = (const float*)d_in[4];
  const float* w2 = (const float*)d_in[5];
  const float* b2 = (const float*)d_in[6];
  const float* w3 = (const float*)d_in[7];
  const float* b3 = (const float*)d_in[8];
  const float* w4 = (const float*)d_in[9];
  const float* b4 = (const float*)d_in[10];
  const float* w5 = (const float*)d_in[11];
  const float* b5 = (const float*)d_in[12];
  const float* w6 = (const float*)d_in[13];
  const float* b6 = (const float*)d_in[14];
  const float* w7 = (const float*)d_in[15];
  const float* b7 = (const float*)d_in[16];
  float* out = (float*)d_out;

  dim3 grid(Nn / NT, Bb);  // 1000 x 4 blocks, each = (b, 16-point tile)
  fused_point_attn<<<grid, NTHREADS, 0, stream>>>(
      xyz, feat, nidx, w1, b1, w2, b2, w3, b3, w4, b4, w5, b5, w6, b6, w7, b7,
      out);
}